// AR_68736656605640
// MI455X (gfx1250) — compile-verified
//
#include <hip/hip_runtime.h>
#include <hip/hip_bf16.h>

typedef float v2f __attribute__((ext_vector_type(2)));
typedef float v4f __attribute__((ext_vector_type(4)));
typedef float v8f __attribute__((ext_vector_type(8)));

// ---------------------------------------------------------------------------
// FP32 GEMM via V_WMMA_F32_16X16X4_F32 (CDNA5, wave32).
// C[M,N] = A * B.  TRANS_A: A[m,k] = Aptr[k*LDA + m], else Aptr[m*LDA + k].
// All strides + K are compile-time so k-offsets fold into load IOFFSETs.
//
// Block = 256 threads = 8 waves arranged 2(M) x 4(N); wave tile = 64x32
// (MT=4 x NT=2 subtiles of 16x16); block tile = 128x128.
// Per K-step (K+=4): 8 WMMAs vs ~8 load instructions.
//
// Fragment layouts (ISA 7.12.2, 32-bit operands, wave32):
//   A 16x4 : lane%16 = M, VGPR v holds K = 2*(lane/16) + v
//   B 4x16 : lane%16 = N, VGPR v holds K = 2*(lane/16) + v
//   D 16x16: VGPR i -> row i + 8*(lane/16), col = lane%16
// ---------------------------------------------------------------------------
template <bool TRANS_A, int K, int LDA, int LDB, int LDC>
__global__ __launch_bounds__(256) void gemm_f32_wmma(
    const float* __restrict__ A, const float* __restrict__ B,
    float* __restrict__ C) {
  constexpr int MT = 4;  // 16-row subtiles per wave
  constexpr int NT = 2;  // 16-col subtiles per wave

  const int lane = threadIdx.x & 31;
  const int wave = threadIdx.x >> 5;
  const int tileM = blockIdx.y * (2 * MT * 16) + (wave & 1) * (MT * 16);
  const int tileN = blockIdx.x * (4 * NT * 16) + (wave >> 1) * (NT * 16);
  const int r = lane & 15;  // M-row of A frag / N-col of B frag
  const int h = lane >> 4;  // which K-pair this half-wave owns

  // Base pointers; each k-iteration advances them by a constant.
  const float* Ap;
  if (TRANS_A) {
    Ap = A + (size_t)(2 * h) * LDA + (tileM + r);  // + 4*LDA per iter
  } else {
    Ap = A + (size_t)(tileM + r) * LDA + 2 * h;    // + 4 per iter
  }
  const float* Bp = B + (size_t)(2 * h) * LDB + (tileN + r);  // + 4*LDB per iter

  v8f acc[MT][NT];
#pragma unroll
  for (int i = 0; i < MT; ++i)
#pragma unroll
    for (int j = 0; j < NT; ++j) acc[i][j] = (v8f){};

#pragma unroll 2
  for (int k = 0; k < K; k += 4) {
    v2f a[MT], b[NT];
#pragma unroll
    for (int i = 0; i < MT; ++i) {
      if (TRANS_A) {
        // immediate offsets: v*LDA + 16*i  (max 3*16 + 1024 floats < 24-bit)
        a[i].x = Ap[16 * i];
        a[i].y = Ap[16 * i + LDA];
      } else {
        // immediate offset: i*16*LDA floats = i*64KB bytes (fits IOFFSET)
        a[i] = *(const v2f*)(Ap + (size_t)i * 16 * LDA);
      }
    }
#pragma unroll
    for (int j = 0; j < NT; ++j) {
      b[j].x = Bp[16 * j];
      b[j].y = Bp[16 * j + LDB];
    }
#pragma unroll
    for (int i = 0; i < MT; ++i)
#pragma unroll
      for (int j = 0; j < NT; ++j)
        acc[i][j] = __builtin_amdgcn_wmma_f32_16x16x4_f32(
            false, a[i], false, b[j], (short)0, acc[i][j], false, false);
    Ap += TRANS_A ? 4 * LDA : 4;
    Bp += 4 * LDB;
  }

  // Store: one base pointer, all-constant immediate offsets.
  float* Cp = C + (size_t)(tileM + 8 * h) * LDC + tileN + r;
#pragma unroll
  for (int i = 0; i < MT; ++i)
#pragma unroll
    for (int j = 0; j < NT; ++j)
#pragma unroll
      for (int v = 0; v < 8; ++v)
        Cp[(size_t)(16 * i + v) * LDC + 16 * j] = acc[i][j][v];
}

// ---------------------------------------------------------------------------
// Fused per-token: scores = c~ . prev_p / sqrt(D), softmax over P=8,
// attn = sum_p w_p * prev_p, y = cur + attn, LayerNorm(y).
// One 256-thread block (8 wave32) per token; each thread owns 4 channels.
// prev is a 268 MB one-shot stream -> non-temporal loads so cur/ctil/out
// (~100 MB combined) stay resident in the 192 MB L2. Deterministic tree
// reductions (shfl_xor within wave, LDS across the 8 waves).
// ---------------------------------------------------------------------------
__global__ __launch_bounds__(256) void fused_attn_ln(
    const float* __restrict__ cur, const float* __restrict__ prev,
    const float* __restrict__ ctil, const float* __restrict__ gamma,
    const float* __restrict__ beta, float* __restrict__ out) {
  constexpr int D = 1024;
  constexpr int P = 8;
  const int token = blockIdx.x;
  const int tid = threadIdx.x;
  const int lane = tid & 31;
  const int wave = tid >> 5;

  const v4f* cur4 = (const v4f*)(cur + (size_t)token * D);
  const v4f* ct4 = (const v4f*)(ctil + (size_t)token * D);
  const v4f* pbase = (const v4f*)(prev + (size_t)token * P * D);

  v4f x = cur4[tid];
  v4f c = ct4[tid];

  v4f pv[P];
  float s[P];
#pragma unroll
  for (int p = 0; p < P; ++p) {
    pv[p] = __builtin_nontemporal_load(pbase + (size_t)p * (D / 4) + tid);
    s[p] = c.x * pv[p].x + c.y * pv[p].y + c.z * pv[p].z + c.w * pv[p].w;
  }

  __shared__ float red[8][P];
  __shared__ float red2[8][2];

#pragma unroll
  for (int p = 0; p < P; ++p) {
#pragma unroll
    for (int off = 16; off >= 1; off >>= 1) s[p] += __shfl_xor(s[p], off, 32);
  }
  if (lane < P) red[wave][lane] = s[lane];
  __syncthreads();

  const float inv_sqrt_d = 0.03125f;  // 1/sqrt(1024)
  float score[P];
#pragma unroll
  for (int p = 0; p < P; ++p) {
    float t = 0.f;
#pragma unroll
    for (int w = 0; w < 8; ++w) t += red[w][p];
    score[p] = t * inv_sqrt_d;
  }

  float mx = score[0];
#pragma unroll
  for (int p = 1; p < P; ++p) mx = fmaxf(mx, score[p]);
  float e[P], esum = 0.f;
#pragma unroll
  for (int p = 0; p < P; ++p) {
    e[p] = __expf(score[p] - mx);
    esum += e[p];
  }
  const float inv = 1.f / esum;

  v4f y = x;
#pragma unroll
  for (int p = 0; p < P; ++p) {
    const float w = e[p] * inv;
    y.x = fmaf(w, pv[p].x, y.x);
    y.y = fmaf(w, pv[p].y, y.y);
    y.z = fmaf(w, pv[p].z, y.z);
    y.w = fmaf(w, pv[p].w, y.w);
  }

  // LayerNorm: single-pass sum & sum-of-squares, tree-reduced.
  float ps = y.x + y.y + y.z + y.w;
  float ps2 = y.x * y.x + y.y * y.y + y.z * y.z + y.w * y.w;
#pragma unroll
  for (int off = 16; off >= 1; off >>= 1) {
    ps += __shfl_xor(ps, off, 32);
    ps2 += __shfl_xor(ps2, off, 32);
  }
  if (lane == 0) {
    red2[wave][0] = ps;
    red2[wave][1] = ps2;
  }
  __syncthreads();
  float S = 0.f, S2 = 0.f;
#pragma unroll
  for (int w = 0; w < 8; ++w) {
    S += red2[w][0];
    S2 += red2[w][1];
  }
  const float mu = S * (1.f / D);
  const float var = S2 * (1.f / D) - mu * mu;
  const float rstd = rsqrtf(var + 1e-5f);

  v4f g = ((const v4f*)gamma)[tid];
  v4f b = ((const v4f*)beta)[tid];
  v4f o;
  o.x = (y.x - mu) * rstd * g.x + b.x;
  o.y = (y.y - mu) * rstd * g.y + b.y;
  o.z = (y.z - mu) * rstd * g.z + b.z;
  o.w = (y.w - mu) * rstd * g.w + b.w;
  __builtin_nontemporal_store(o, (v4f*)(out + (size_t)token * D) + tid);
}

extern "C" void kernel_launch(void* const* d_in, const int* in_sizes, int n_in,
                              void* d_out, int out_size, void* d_ws,
                              size_t ws_size, hipStream_t stream) {
  const float* cur = (const float*)d_in[0];    // [B,T,D]
  const float* prev = (const float*)d_in[1];   // [B,T,P,D]
  const float* Wq = (const float*)d_in[2];     // [D,D]
  const float* Wk = (const float*)d_in[3];     // [D,D]
  const float* gamma = (const float*)d_in[4];  // [D]
  const float* beta = (const float*)d_in[5];   // [D]
  float* out = (float*)d_out;

  constexpr int D = 1024;
  const int M = in_sizes[0] / D;  // B*T = 8192 tokens

  float* G = (float*)d_ws;          // [D,D] = Wq^T * Wk
  float* Ctil = G + (size_t)D * D;  // [M,D] = cur * G

  dim3 blk(256);

  // 1) G = Wq^T @ Wk  (A[m,k] = Wq[k,m] -> TRANS_A). Block tile 128x128.
  gemm_f32_wmma<true, D, D, D, D>
      <<<dim3(D / 128, D / 128), blk, 0, stream>>>(Wq, Wk, G);

  // 2) Ctil = cur @ G
  gemm_f32_wmma<false, D, D, D, D>
      <<<dim3(D / 128, M / 128), blk, 0, stream>>>(cur, G, Ctil);

  // 3) fused scores/softmax/attn/residual/LayerNorm, one block per token
  fused_attn_ln<<<dim3(M), blk, 0, stream>>>(cur, prev, Ctil, gamma, beta, out);
}